// SplitWin_Normalize_62139586838835
// MI455X (gfx1250) — compile-verified
//
#include <hip/hip_runtime.h>
#include <cstdint>

// Problem geometry (reference: signal (1, 4096, 4096) fp32; filter axis = slow axis)
#define NBINS 4096
#define BLK   128     // threads per block = 4 wave32
#define VEC   4       // lines per thread (float4 / b128 path)
#define SEG   32      // bin segments per line (grid: 8 x 32 blocks = 1024 waves)
#define RING  32      // LDS ring slots per thread (must be > PIPE)
#define PIPE  16      // async loads in flight per thread

typedef float v4f __attribute__((ext_vector_type(4)));

// numpy 'symmetric' pad index map, valid for j in [-n, 2n)
__device__ __forceinline__ int refl(int j, int n) {
    if (j < 0)  return -1 - j;
    if (j >= n) return 2 * n - 1 - j;
    return j;
}

// CDNA5 async global->LDS, 16B per lane (GVS mode: saddr base + 32-bit vaddr
// byte offset). Tracked by ASYNCcnt; per-lane LDS dest address in a VGPR.
__device__ __forceinline__ void async_ld_b128(uint32_t lds_off, const float* base, uint32_t byte_off) {
    asm volatile("global_load_async_to_lds_b128 %0, %1, %2"
                 :: "v"(lds_off), "v"(byte_off), "s"(base)
                 : "memory");
}

__device__ __forceinline__ void wait_async_le_pipe() {
    asm volatile("s_wait_asynccnt %0" :: "i"(PIPE - 1) : "memory");
}

// Box filter of width SIZE along bins (slow axis), symmetric padding.
// Each thread owns VEC=4 consecutive lines and a contiguous bin segment;
// vector running-window sum. Incoming window element (16B) is fetched PIPE
// bins ahead via async->LDS ring. If FDIV: dst = xin / filtered.
template <int SIZE, bool FDIV>
__global__ void filt_kernel(const float* __restrict__ src,
                            const float* __restrict__ xin,
                            float* __restrict__ dst,
                            int nlines) {
    constexpr int   LEFT  = SIZE / 2;
    constexpr int   RIGHT = SIZE - LEFT - 1;
    constexpr float INV   = 1.0f / (float)SIZE;
    constexpr int   SLEN  = NBINS / SEG;

    const int l4 = (blockIdx.x * BLK + threadIdx.x) * VEC;  // first of 4 lines
    const int s0 = blockIdx.y * SLEN;                       // first output bin
    const int s1 = s0 + SLEN;

    __shared__ v4f ring[RING * BLK];
    const uint32_t lds_base = (uint32_t)(uintptr_t)(&ring[threadIdx.x]);

    // Initial window sum for output bin s0 (reflected indices; L2 hits).
    v4f sum = (v4f)0.0f;
    for (int t = -LEFT; t <= RIGHT; ++t)
        sum += *(const v4f*)(src + (size_t)refl(s0 + t, NBINS) * nlines + l4);

    // Prime the pipeline: incoming(j) = src[refl(j + 1 + RIGHT)]
    for (int j = s0; j < s0 + PIPE; ++j) {
        uint32_t off = ((uint32_t)refl(j + 1 + RIGHT, NBINS) * (uint32_t)nlines + (uint32_t)l4) * 4u;
        async_ld_b128(lds_base + (uint32_t)(j & (RING - 1)) * (BLK * 16u), src, off);
    }

    for (int j = s0; j < s1; ++j) {
        const v4f f = sum * INV;
        const size_t o = (size_t)j * nlines + l4;
        if (FDIV) {
            const v4f xv = *(const v4f*)(xin + o);
            *(v4f*)(dst + o) = xv / f;
        } else {
            *(v4f*)(dst + o) = f;
        }

        // Keep exactly PIPE loads outstanding (tail issues reflected-but-valid
        // addresses so the constant s_wait_asynccnt PIPE-1 stays sufficient;
        // write slot is PIPE < RING ahead of the read slot -> no collision).
        const int jp = j + PIPE;
        {
            uint32_t off = ((uint32_t)refl(jp + 1 + RIGHT, NBINS) * (uint32_t)nlines + (uint32_t)l4) * 4u;
            async_ld_b128(lds_base + (uint32_t)(jp & (RING - 1)) * (BLK * 16u), src, off);
        }
        wait_async_le_pipe();  // ASYNCcnt completes in order -> incoming(j) landed

        const v4f inc  = ring[(j & (RING - 1)) * BLK + threadIdx.x];
        const v4f outg = *(const v4f*)(src + (size_t)refl(j - LEFT, NBINS) * nlines + l4);  // L2 hit
        sum += inc - outg;
    }
}

// Element-wise clip stage: smoothed = (filt[fwd] + filt[aft]) / 2,
// clipped' = (clipped > clip*smoothed) ? smoothed : clipped.
// winsz=41, gap=20 for both gap stages: fwd = i+41 (reflect), aft = floor(|i-41.5|).
__global__ void clip_kernel(const float* __restrict__ cin,
                            const float* __restrict__ filt,
                            float* __restrict__ cout,
                            int nlines, float clipv) {
    const int l4 = (blockIdx.x * blockDim.x + threadIdx.x) * VEC;  // lines -> coalesced b128
    const int b  = blockIdx.y;                                     // bin
    int fwd = b + 41; if (fwd >= NBINS) fwd = 2 * NBINS - 2 - fwd;
    const int aft = (b >= 42) ? (b - 42) : (41 - b);

    const v4f ff = *(const v4f*)(filt + (size_t)fwd * nlines + l4);
    const v4f fa = *(const v4f*)(filt + (size_t)aft * nlines + l4);
    const v4f sm = 0.5f * (ff + fa);
    const size_t g = (size_t)b * nlines + l4;
    const v4f c = *(const v4f*)(cin + g);
    v4f r;
    #pragma unroll
    for (int k = 0; k < VEC; ++k)
        r[k] = (c[k] > clipv * sm[k]) ? sm[k] : c[k];
    *(v4f*)(cout + g) = r;
}

extern "C" void kernel_launch(void* const* d_in, const int* in_sizes, int n_in,
                              void* d_out, int out_size, void* d_ws, size_t ws_size,
                              hipStream_t stream) {
    const float* x = (const float*)d_in[0];   // (NBINS, nlines) row-major, fp32
    float* out = (float*)d_out;

    const int nlines = in_sizes[0] / NBINS;   // 4096
    const size_t plane = (size_t)NBINS * (size_t)nlines;

    float* A = (float*)d_ws;                  // filtered scratch (64 MB)
    float* B = A + plane;                     // clipped  scratch (64 MB)

    dim3 fgrid(nlines / (BLK * VEC), SEG), fblk(BLK);
    dim3 cgrid(nlines / (256 * VEC), NBINS), cblk(256);

    // Stage 1: f1 = box41(x);  c1 = clip(x, f1, 3)
    filt_kernel<41, false><<<fgrid, fblk, 0, stream>>>(x, nullptr, A, nlines);
    clip_kernel<<<cgrid, cblk, 0, stream>>>(x, A, B, nlines, 3.0f);
    // Stage 2: f2 = box41(c1); c2 = clip(c1, f2, 2)  (clip reads its own index only -> in-place safe)
    filt_kernel<41, false><<<fgrid, fblk, 0, stream>>>(B, nullptr, A, nlines);
    clip_kernel<<<cgrid, cblk, 0, stream>>>(B, A, B, nlines, 2.0f);
    // Stage 3 + normalize: out = x / box25(c2)
    filt_kernel<25, true><<<fgrid, fblk, 0, stream>>>(B, x, out, nlines);
}